// GraphDiffusionLayer_7937099563713
// MI455X (gfx1250) — compile-verified
//
#include <hip/hip_runtime.h>
#include <hip/hip_bf16.h>

typedef __attribute__((ext_vector_type(2))) float v2f;
typedef __attribute__((ext_vector_type(8))) float v8f;

#define NN      20000
#define EE      320000
#define FF      512
#define NSTEP   5
#define DT      (1.0f / 5.0f)     // DIFF_T / NUM_STEPS
#define DECAY   0.1f
#define LNEPS   1e-5f

// ---------------------------------------------------------------------------
// 1) edge weights + degree accumulation (self-loops appended at [E, E+N))
// ---------------------------------------------------------------------------
__global__ void edge_w_deg_kernel(const int* __restrict__ ei,
                                  const float* __restrict__ ts,
                                  const int* __restrict__ gtime,
                                  float* __restrict__ w,
                                  float* __restrict__ deg) {
    int i = blockIdx.x * blockDim.x + threadIdx.x;
    const int EN = EE + NN;
    if (i >= EN) return;
    float tmax = (float)(*gtime);          // ts.max() == graph_time (ts < 100 strictly)
    int d; float t;
    if (i < EE) { d = ei[EE + i]; t = ts[i]; }
    else        { d = i - EE;     t = tmax; }
    float wi = expf(-DECAY * (tmax - t));
    w[i] = wi;
    unsafeAtomicAdd(&deg[d], wi);
}

// deg -> deg^-0.5 in place
__global__ void dinv_kernel(float* __restrict__ deg) {
    int i = blockIdx.x * blockDim.x + threadIdx.x;
    if (i >= NN) return;
    float v = deg[i];
    deg[i] = (v > 0.0f) ? (1.0f / sqrtf(v)) : 0.0f;
}

// w -> dinv[src]*w*dinv[dst]*dt in place
__global__ void ew_kernel(const int* __restrict__ ei,
                          const float* __restrict__ dinv,
                          float* __restrict__ ew) {
    int i = blockIdx.x * blockDim.x + threadIdx.x;
    const int EN = EE + NN;
    if (i >= EN) return;
    int s, d;
    if (i < EE) { s = ei[i]; d = ei[EE + i]; }
    else        { s = d = i - EE; }
    ew[i] = dinv[s] * ew[i] * dinv[d] * DT;
}

// h1 = (1-dt) * h0   (float4 vectorized; NN*FF divisible by 4)
__global__ void init_step_kernel(const float4* __restrict__ hin,
                                 float4* __restrict__ hout) {
    int i = blockIdx.x * blockDim.x + threadIdx.x;
    if (i < NN * FF / 4) {
        float4 v = hin[i];
        const float c = 1.0f - DT;
        v.x *= c; v.y *= c; v.z *= c; v.w *= c;
        hout[i] = v;
    }
}

// scatter: h1[dst] += ew'[e] * h0[src] ; one block per edge, 256 thr x 2 feats
__global__ void scatter_kernel(const int* __restrict__ ei,
                               const float* __restrict__ ew,
                               const float* __restrict__ hin,
                               float* __restrict__ hout) {
    int e = blockIdx.x;
    int s, d;
    if (e < EE) { s = ei[e]; d = ei[EE + e]; }
    else        { s = d = e - EE; }
    float we = ew[e];
    const float* __restrict__ src = hin + (size_t)s * FF;
    float* __restrict__ dst = hout + (size_t)d * FF;
    int f = threadIdx.x;
    unsafeAtomicAdd(&dst[f],       we * src[f]);
    unsafeAtomicAdd(&dst[f + 256], we * src[f + 256]);
}

// ---------------------------------------------------------------------------
// 2) fused dual GEMM: pre = relu(h @ Wt^T + bt) + x @ Wr^T
//    block tile 64(M) x 64(N); K staged in 32-wide chunks via
//    GLOBAL_LOAD_ASYNC_TO_LDS_B128, double-buffered (asynccnt pipeline);
//    8 waves = 4 Mtiles x 2 Nhalves, v_wmma_f32_16x16x4_f32 accumulation
// ---------------------------------------------------------------------------
#define LDP  36                 // pitch: conflict-free (gcd trick) AND 16B-aligned rows
#define TILE (64 * LDP)         // floats per staged array

__device__ __forceinline__ void async_b128(const float* gsrc, float* ldst) {
    unsigned la = (unsigned)(uintptr_t)ldst;   // low 32 bits of generic = LDS byte addr
    asm volatile("global_load_async_to_lds_b128 %0, %1, off"
                 :: "v"(la), "v"(gsrc) : "memory");
}

__global__ __launch_bounds__(256)
void gemm_fused_kernel(const float* __restrict__ H,   // diffused h [N,512]
                       const float* __restrict__ X,   // residual x [N,512]
                       const float* __restrict__ Wt,  // [512,512] row = out col
                       const float* __restrict__ Bt,  // [512]
                       const float* __restrict__ Wr,  // [512,512]
                       float* __restrict__ pre) {     // [N,512] pre-layernorm
    __shared__ __align__(16) float smem[2][4][TILE];  // [buf][h,x,wt,wr][64x36]

    const int tid  = threadIdx.x;
    const int lane = tid & 31;
    const int wid  = tid >> 5;
    const int wm   = wid & 3;        // M sub-tile (16 rows)
    const int wn   = wid >> 2;       // N half (32 cols)
    const int lm   = lane & 15;
    const int hi   = lane >> 4;      // 0: lanes 0-15, 1: lanes 16-31
    const int M0   = blockIdx.x * 64;
    const int N0   = blockIdx.y * 64;

    v8f acc_t[2] = {};
    v8f acc_r[2] = {};

    // staging geometry: 64x32 tile = 512 float4; thread does 2 float4 per array
    const int r0  = (tid      ) >> 3;          // row for q=0  (e4 = tid)
    const int c40 = (tid      ) &  7;
    const int r1  = (tid + 256) >> 3;          // row for q=1
    const int c41 = (tid + 256) &  7;

    auto stage = [&](int k0, int buf) {
        {
            int gr = M0 + r0; if (gr >= NN) gr = NN - 1;   // clamp: junk rows never stored
            int wr_ = N0 + r0;
            size_t go = (size_t)gr * FF + k0 + c40 * 4;
            size_t wo = (size_t)wr_ * FF + k0 + c40 * 4;
            int lo = r0 * LDP + c40 * 4;
            async_b128(H  + go, &smem[buf][0][lo]);
            async_b128(X  + go, &smem[buf][1][lo]);
            async_b128(Wt + wo, &smem[buf][2][lo]);
            async_b128(Wr + wo, &smem[buf][3][lo]);
        }
        {
            int gr = M0 + r1; if (gr >= NN) gr = NN - 1;
            int wr_ = N0 + r1;
            size_t go = (size_t)gr * FF + k0 + c41 * 4;
            size_t wo = (size_t)wr_ * FF + k0 + c41 * 4;
            int lo = r1 * LDP + c41 * 4;
            async_b128(H  + go, &smem[buf][0][lo]);
            async_b128(X  + go, &smem[buf][1][lo]);
            async_b128(Wt + wo, &smem[buf][2][lo]);
            async_b128(Wr + wo, &smem[buf][3][lo]);
        }
    };

    stage(0, 0);                                   // prologue: tile 0 in flight

    const int NK = FF / 32;                        // 16 k-chunks
    for (int it = 0; it < NK; ++it) {
        int cur = it & 1;
        if (it + 1 < NK) {
            stage((it + 1) * 32, cur ^ 1);         // prefetch next tile (8 async ops)
            asm volatile("s_wait_asynccnt 8" ::: "memory");   // tile `it` complete
        } else {
            asm volatile("s_wait_asynccnt 0" ::: "memory");
        }
        __syncthreads();                           // all waves' loads visible

        const float* s_h  = smem[cur][0];
        const float* s_x  = smem[cur][1];
        const float* s_wt = smem[cur][2];
        const float* s_wr = smem[cur][3];

        #pragma unroll
        for (int kk = 0; kk < 32; kk += 4) {
            int kb = kk + (hi << 1);               // low lanes K={0,1}, high K={2,3}
            int ar = (wm * 16 + lm) * LDP + kb;
            v2f ah; ah.x = s_h[ar]; ah.y = s_h[ar + 1];
            v2f ax; ax.x = s_x[ar]; ax.y = s_x[ar + 1];
            #pragma unroll
            for (int j = 0; j < 2; ++j) {
                int bc = (wn * 32 + j * 16 + lm) * LDP + kb;
                v2f bt_; bt_.x = s_wt[bc]; bt_.y = s_wt[bc + 1];
                v2f br_; br_.x = s_wr[bc]; br_.y = s_wr[bc + 1];
                acc_t[j] = __builtin_amdgcn_wmma_f32_16x16x4_f32(
                    false, ah, false, bt_, (short)0, acc_t[j], false, false);
                acc_r[j] = __builtin_amdgcn_wmma_f32_16x16x4_f32(
                    false, ax, false, br_, (short)0, acc_r[j], false, false);
            }
        }
        __syncthreads();                           // done reading before overwrite
    }

    // epilogue: D layout — VGPR v holds row v (+8 for high lanes), col = lane%16
    #pragma unroll
    for (int j = 0; j < 2; ++j) {
        int col = N0 + wn * 32 + j * 16 + lm;
        float bias = Bt[col];
        #pragma unroll
        for (int v = 0; v < 8; ++v) {
            int row = M0 + wm * 16 + v + (hi << 3);
            if (row < NN) {
                float tv = acc_t[j][v] + bias;
                tv = tv > 0.0f ? tv : 0.0f;
                pre[(size_t)row * FF + col] = tv + acc_r[j][v];
            }
        }
    }
}

// ---------------------------------------------------------------------------
// 3) layernorm: one block per row of 512
// ---------------------------------------------------------------------------
__global__ __launch_bounds__(256)
void layernorm_kernel(const float* __restrict__ pre,
                      const float* __restrict__ gamma,
                      const float* __restrict__ beta,
                      float* __restrict__ out) {
    __shared__ float ssum[256];
    __shared__ float ssq[256];
    int row = blockIdx.x;
    int tid = threadIdx.x;
    const float* p = pre + (size_t)row * FF;
    float v0 = p[tid], v1 = p[tid + 256];
    ssum[tid] = v0 + v1;
    ssq[tid]  = v0 * v0 + v1 * v1;
    __syncthreads();
    #pragma unroll
    for (int o = 128; o > 0; o >>= 1) {
        if (tid < o) { ssum[tid] += ssum[tid + o]; ssq[tid] += ssq[tid + o]; }
        __syncthreads();
    }
    float mu  = ssum[0] * (1.0f / FF);
    float var = ssq[0] * (1.0f / FF) - mu * mu;
    float rs  = rsqrtf(var + LNEPS);
    out[(size_t)row * FF + tid]       = (v0 - mu) * rs * gamma[tid]       + beta[tid];
    out[(size_t)row * FF + tid + 256] = (v1 - mu) * rs * gamma[tid + 256] + beta[tid + 256];
}

// ---------------------------------------------------------------------------
static inline size_t align256(size_t x) { return (x + 255) & ~(size_t)255; }

extern "C" void kernel_launch(void* const* d_in, const int* in_sizes, int n_in,
                              void* d_out, int out_size, void* d_ws, size_t ws_size,
                              hipStream_t stream) {
    const float* x     = (const float*)d_in[0];
    const int*   ei    = (const int*)  d_in[1];
    const float* ts    = (const float*)d_in[2];
    const float* Wt    = (const float*)d_in[3];
    const float* bt    = (const float*)d_in[4];
    const float* Wr    = (const float*)d_in[5];
    const float* gamma = (const float*)d_in[6];
    const float* beta  = (const float*)d_in[7];
    const int*   gtime = (const int*)  d_in[8];
    float* out = (float*)d_out;

    const int EN = EE + NN;
    char* ws = (char*)d_ws;
    size_t off = 0;
    float* ew  = (float*)(ws + off); off = align256(off + (size_t)EN * 4);
    float* deg = (float*)(ws + off); off = align256(off + (size_t)NN * 4);
    float* ha  = (float*)(ws + off); off = align256(off + (size_t)NN * FF * 4);
    float* hb  = (float*)(ws + off); off = align256(off + (size_t)NN * FF * 4);

    // edge weights + degrees
    hipMemsetAsync(deg, 0, (size_t)NN * 4, stream);
    edge_w_deg_kernel<<<(EN + 255) / 256, 256, 0, stream>>>(ei, ts, gtime, ew, deg);
    dinv_kernel<<<(NN + 255) / 256, 256, 0, stream>>>(deg);
    ew_kernel<<<(EN + 255) / 256, 256, 0, stream>>>(ei, deg, ew);

    // 5 Euler steps: x -> ha -> hb -> ha -> hb -> ha
    const float* hin = x;
    float* hout = ha;
    for (int s = 0; s < NSTEP; ++s) {
        init_step_kernel<<<(NN * FF / 4 + 255) / 256, 256, 0, stream>>>(
            (const float4*)hin, (float4*)hout);
        scatter_kernel<<<EN, 256, 0, stream>>>(ei, ew, hin, hout);
        hin  = hout;
        hout = (hout == ha) ? hb : ha;
    }
    const float* hfinal = hin;          // == ha after 5 steps
    float* pre = (hfinal == ha) ? hb : ha;

    // fused dual GEMM with f32 WMMA + async-to-LDS double buffering
    dim3 ggrid((NN + 63) / 64, FF / 64);
    gemm_fused_kernel<<<ggrid, 256, 0, stream>>>(hfinal, x, Wt, bt, Wr, pre);

    // layernorm
    layernorm_kernel<<<NN, 256, 0, stream>>>(pre, gamma, beta, out);
}